// HeatEquationOperators_52536039965342
// MI455X (gfx1250) — compile-verified
//
#include <hip/hip_runtime.h>
#include <hip/hip_bf16.h>

typedef __attribute__((ext_vector_type(2))) float v2f;
typedef __attribute__((ext_vector_type(8))) float v8f;

#define ALPHA_C 0.01f
#define ROWS_PER_BLOCK 16
#define COLS_PER_BLOCK 128   // 8 waves * 16 cols each

struct HeatConsts {
    float gam;    // γ = 1/ℓ²
    float c_ag;   // αγ
    float c_ag2;  // αγ²
    float c0c;    // γ + 2α²γ²
    float c1c;    // 4α²γ³
    float nhg;    // -γ/2
    float sig2;   // σ²
};

// Closed form: with u = x1-x2, f = σ² exp(-γ r²/2),
//   A = γu1 + αγ − αγ²u0²
//   out = f · [ γ + 2α²γ² − γu1A − A(αγ²u0² − αγ) − 4α²γ³u0² ]
__device__ __forceinline__ float heat_val(float dot, float n1, float ncol,
                                          float u1, const HeatConsts& c) {
    const float r2   = n1 + ncol - 2.f * dot;       // |x1-x2|^2 (WMMA cross term)
    float u0sq = r2 - u1 * u1;
    u0sq = fmaxf(u0sq, 0.f);
    const float Av = fmaf(c.gam, u1, c.c_ag - c.c_ag2 * u0sq);
    const float P  = c.c0c - c.gam * u1 * Av - Av * (c.c_ag2 * u0sq - c.c_ag)
                     - c.c1c * u0sq;
    return c.sig2 * __expf(c.nhg * r2) * P;
}

// N2C > 0: compile-time column count (immediate store offsets); N2C == 0: runtime.
template <int N2C>
__global__ __launch_bounds__(256)
void HeatEquationOperators_52536039965342_kernel(
    const float* __restrict__ X1, const float* __restrict__ X2,
    const float* __restrict__ lsp, const float* __restrict__ s2p,
    float* __restrict__ out, int N1, int N2rt)
{
    const int N2 = (N2C > 0) ? N2C : N2rt;

    __shared__ float r0s[ROWS_PER_BLOCK];   // X1 space coord
    __shared__ float t1s[ROWS_PER_BLOCK];   // X1 time coord
    __shared__ float n1s[ROWS_PER_BLOCK];   // |x1|^2
    __shared__ float c0s[COLS_PER_BLOCK];   // X2 space coord
    __shared__ float t2s[COLS_PER_BLOCK];   // X2 time coord
    __shared__ float n2s[COLS_PER_BLOCK];   // |x2|^2

    const int tid      = threadIdx.x;
    const int rowBase  = blockIdx.y * ROWS_PER_BLOCK;
    const int colBlock = blockIdx.x * COLS_PER_BLOCK;

    // Stage the (tiny) coordinate tiles into LDS once per block.
    if (tid < ROWS_PER_BLOCK) {
        int r = rowBase + tid;
        float x0 = 0.f, x1t = 0.f;
        if (r < N1) { x0 = X1[2 * r]; x1t = X1[2 * r + 1]; }
        r0s[tid] = x0; t1s[tid] = x1t; n1s[tid] = x0 * x0 + x1t * x1t;
    }
    if (tid < COLS_PER_BLOCK) {
        int c = colBlock + tid;
        float y0 = 0.f, y1t = 0.f;
        if (c < N2) { y0 = X2[2 * c]; y1t = X2[2 * c + 1]; }
        c0s[tid] = y0; t2s[tid] = y1t; n2s[tid] = y0 * y0 + y1t * y1t;
    }
    __syncthreads();

    // Uniform per-thread constants.
    HeatConsts hc;
    {
        const float ell = lsp[0];
        const float g   = 1.0f / (ell * ell);
        hc.gam  = g;
        hc.c_ag  = ALPHA_C * g;
        hc.c_ag2 = ALPHA_C * g * g;
        hc.c0c   = g + 2.f * ALPHA_C * ALPHA_C * g * g;
        hc.c1c   = 4.f * ALPHA_C * ALPHA_C * g * g * g;
        hc.nhg   = -0.5f * g;
        hc.sig2  = s2p[0];
    }

    const int wave = tid >> 5;
    const int lane = tid & 31;
    const int hi   = lane >> 4;   // half-wave select (rows +8)
    const int ln   = lane & 15;
    const int col  = colBlock + wave * 16 + ln;

    // Per-lane column values (fixed for the whole tile).
    const float tcol = t2s[wave * 16 + ln];
    const float ncol = n2s[wave * 16 + ln];

    // Hoist the 8 per-row LDS values into registers (broadcast reads).
    float t1r[8], n1r[8];
#pragma unroll
    for (int v = 0; v < 8; ++v) {
        const int m = v + 8 * hi;
        t1r[v] = t1s[m];
        n1r[v] = n1s[m];
    }

    // A (16x4 f32): lanes 0-15 carry K=0,1 = (x, t); lanes 16-31 carry K=2,3 = 0.
    v2f a, b;
    if (hi == 0) {
        a[0] = r0s[ln];             a[1] = t1s[ln];
        b[0] = c0s[wave * 16 + ln]; b[1] = t2s[wave * 16 + ln];
    } else {
        a[0] = 0.f; a[1] = 0.f; b[0] = 0.f; b[1] = 0.f;
    }
    v8f acc = {};
    // acc(M,N) = x1[rowBase+M] . x2[colBase+N]
    acc = __builtin_amdgcn_wmma_f32_16x16x4_f32(
        /*neg_a=*/false, a, /*neg_b=*/false, b,
        /*c_mod=*/(short)0, acc, /*reuse_a=*/false, /*reuse_b=*/false);

    // Block-uniform interior test: hot path has no per-element guards.
    const bool interior = (rowBase + ROWS_PER_BLOCK <= N1) &&
                          (colBlock + COLS_PER_BLOCK <= N2);

    if (interior) {
        // One base address per lane; rows at constant (or uniform) offsets.
        float* __restrict__ p =
            out + (size_t)(rowBase + 8 * hi) * (size_t)N2 + (size_t)col;
#pragma unroll
        for (int v = 0; v < 8; ++v) {
            const float u1  = t1r[v] - tcol;
            const float val = heat_val(acc[v], n1r[v], ncol, u1, hc);
            p[(size_t)v * (size_t)N2] = val;   // N2C: immediate offsets
        }
    } else {
#pragma unroll
        for (int v = 0; v < 8; ++v) {
            const int row = rowBase + v + 8 * hi;
            const float u1  = t1r[v] - tcol;
            const float val = heat_val(acc[v], n1r[v], ncol, u1, hc);
            if (row < N1 && col < N2)
                out[(size_t)row * (size_t)N2 + (size_t)col] = val;
        }
    }
}

extern "C" void kernel_launch(void* const* d_in, const int* in_sizes, int n_in,
                              void* d_out, int out_size, void* d_ws, size_t ws_size,
                              hipStream_t stream) {
    const float* X1 = (const float*)d_in[0];
    const float* X2 = (const float*)d_in[1];
    const float* ls = (const float*)d_in[2];
    const float* s2 = (const float*)d_in[3];
    float* out = (float*)d_out;

    const int N1 = in_sizes[0] / 2;
    const int N2 = in_sizes[1] / 2;

    dim3 grid((N2 + COLS_PER_BLOCK - 1) / COLS_PER_BLOCK,
              (N1 + ROWS_PER_BLOCK - 1) / ROWS_PER_BLOCK);
    if (N2 == 2048) {
        HeatEquationOperators_52536039965342_kernel<2048><<<grid, 256, 0, stream>>>(
            X1, X2, ls, s2, out, N1, N2);
    } else {
        HeatEquationOperators_52536039965342_kernel<0><<<grid, 256, 0, stream>>>(
            X1, X2, ls, s2, out, N1, N2);
    }
}